// LightGCN_75685913690231
// MI455X (gfx1250) — compile-verified
//
#include <hip/hip_runtime.h>
#include <hip/hip_bf16.h>

#define USERS_COUNT 100000
#define ITEMS_COUNT 50000
#define LATENT_DIM  64
#define N_LAYERS    3
#define NNZ_COUNT   4800000
#define N_NODES     (USERS_COUNT + ITEMS_COUNT)
#define BATCH_USERS 1024
#define EMB_ELEMS   (N_NODES * LATENT_DIM)   // 9,600,000 floats per buffer

typedef float v2f __attribute__((ext_vector_type(2)));
typedef float v8f __attribute__((ext_vector_type(8)));

// ---------------------------------------------------------------------------
// concat user_emb+item_emb into cur, zero the acc buffer (float4 vectorized)
// ---------------------------------------------------------------------------
__global__ __launch_bounds__(256) void lgcn_init_kernel(
    const float* __restrict__ user_emb,
    const float* __restrict__ item_emb,
    float* __restrict__ cur,
    float* __restrict__ acc)
{
    int i4 = blockIdx.x * blockDim.x + threadIdx.x;          // float4 index
    if (i4 >= EMB_ELEMS / 4) return;
    int i = i4 * 4;
    float4 v;
    if (i < USERS_COUNT * LATENT_DIM) {
        v = *(const float4*)(user_emb + i);
    } else {
        v = *(const float4*)(item_emb + (i - USERS_COUNT * LATENT_DIM));
    }
    *(float4*)(cur + i) = v;
    *(float4*)(acc + i) = make_float4(0.f, 0.f, 0.f, 0.f);
}

__global__ __launch_bounds__(256) void lgcn_zero_kernel(float* __restrict__ p)
{
    int i4 = blockIdx.x * blockDim.x + threadIdx.x;
    if (i4 >= EMB_ELEMS / 4) return;
    *(float4*)(p + i4 * 4) = make_float4(0.f, 0.f, 0.f, 0.f);
}

// ---------------------------------------------------------------------------
// SpMM: y[rows[e]] += x[cols[e]] * vals[e]    (COO, atomic scatter)
// 16 threads per edge, 4 floats (one float4 gather) per thread.
// Working set (x + y ~= 77 MB) fits in the 192 MB L2, so the atomics and
// gathers resolve at L2 bandwidth, not HBM.
// ---------------------------------------------------------------------------
__global__ __launch_bounds__(256) void lgcn_spmm_kernel(
    const int*   __restrict__ rows,
    const int*   __restrict__ cols,
    const float* __restrict__ vals,
    const float* __restrict__ x,
    float*       __restrict__ y)
{
    int tid  = blockIdx.x * blockDim.x + threadIdx.x;
    int e    = tid >> 4;                 // edge index
    if (e >= NNZ_COUNT) return;
    int part = (tid & 15) * 4;           // dim offset 0..60

    int   r = rows[e];
    int   c = cols[e];
    float v = vals[e];

    const float4 s = *(const float4*)(x + (size_t)c * LATENT_DIM + part);
    float* dst = y + (size_t)r * LATENT_DIM + part;
    atomicAdd(dst + 0, s.x * v);
    atomicAdd(dst + 1, s.y * v);
    atomicAdd(dst + 2, s.z * v);
    atomicAdd(dst + 3, s.w * v);
}

// acc += next (float4 vectorized)
__global__ __launch_bounds__(256) void lgcn_acc_kernel(
    float* __restrict__ acc, const float* __restrict__ nxt)
{
    int i4 = blockIdx.x * blockDim.x + threadIdx.x;
    if (i4 >= EMB_ELEMS / 4) return;
    int i = i4 * 4;
    float4 a = *(const float4*)(acc + i);
    float4 n = *(const float4*)(nxt + i);
    a.x += n.x; a.y += n.y; a.z += n.z; a.w += n.w;
    *(float4*)(acc + i) = a;
}

// ---------------------------------------------------------------------------
// rating = sigmoid( (acc[user_ids] @ acc[items]^T) / 9 )
// One wave32 per 16x16 output tile, K=64 via 16x V_WMMA_F32_16X16X4_F32.
// A (16x4 f32): lanes 0-15 hold M=0..15 with K={0,1} in the two VGPRs,
//               lanes 16-31 hold K={2,3}            (ISA 7.12.2)
// B (4x16 f32): lanes 0-15 hold N=0..15 with K={0,1}, lanes 16-31 K={2,3}
// C/D (16x16 f32): VGPR r, lane l<16 -> C[r][l]; lane>=16 -> C[8+r][l-16]
// ---------------------------------------------------------------------------
__global__ __launch_bounds__(256) void lgcn_gemm_sigmoid_kernel(
    const float* __restrict__ acc,        // N_NODES x 64 (un-divided sum)
    const int*   __restrict__ user_ids,   // BATCH_USERS
    float*       __restrict__ out)        // BATCH_USERS x ITEMS_COUNT
{
    const int TILES_N = ITEMS_COUNT / 16;                       // 3125
    int wave = blockIdx.x * (blockDim.x >> 5) + (threadIdx.x >> 5);
    int tile_m = wave / TILES_N;
    int tile_n = wave % TILES_N;
    if (tile_m >= BATCH_USERS / 16) return;                     // wave-uniform

    int lane = threadIdx.x & 31;
    int half = lane >> 4;            // 0: K={0,1}   1: K={2,3}
    int l15  = lane & 15;
    int koff = half * 2;

    // A operand: this lane walks one user row
    int urow = user_ids[tile_m * 16 + l15];
    const float* aptr = acc + (size_t)urow * LATENT_DIM + koff;
    // B operand: this lane walks one item row (B[k][n] = Items[n][k])
    const float* bptr = acc + ((size_t)USERS_COUNT + tile_n * 16 + l15) * LATENT_DIM + koff;

    v8f c = {0.f, 0.f, 0.f, 0.f, 0.f, 0.f, 0.f, 0.f};
#pragma unroll
    for (int k = 0; k < LATENT_DIM; k += 4) {
        v2f a = *(const v2f*)(aptr + k);
        v2f b = *(const v2f*)(bptr + k);
        c = __builtin_amdgcn_wmma_f32_16x16x4_f32(
                /*neg_a=*/false, a, /*neg_b=*/false, b,
                /*c_mod=*/(short)0, c, /*reuse_a=*/false, /*reuse_b=*/false);
    }

    // (users/3)·(items/3) => scale dot by 1/9, then sigmoid
    const float scale = 1.0f / 9.0f;
    int m_base = tile_m * 16 + half * 8;
    int n      = tile_n * 16 + l15;
#pragma unroll
    for (int r = 0; r < 8; ++r) {
        float xv = c[r] * scale;
        out[(size_t)(m_base + r) * ITEMS_COUNT + n] = 1.0f / (1.0f + __expf(-xv));
    }
}

// ---------------------------------------------------------------------------
extern "C" void kernel_launch(void* const* d_in, const int* in_sizes, int n_in,
                              void* d_out, int out_size, void* d_ws, size_t ws_size,
                              hipStream_t stream)
{
    const float* user_emb = (const float*)d_in[0];
    const float* item_emb = (const float*)d_in[1];
    const float* vals     = (const float*)d_in[2];
    const int*   rows     = (const int*)d_in[3];
    const int*   cols     = (const int*)d_in[4];
    const int*   user_ids = (const int*)d_in[5];
    float*       out      = (float*)d_out;

    float* emb0 = (float*)d_ws;               // 9.6M floats
    float* emb1 = emb0 + EMB_ELEMS;           // 9.6M floats
    float* accb = emb1 + EMB_ELEMS;           // 9.6M floats

    const int vecN   = EMB_ELEMS / 4;                         // 2.4M float4s
    const int vecGrid = (vecN + 255) / 256;
    const int spmmGrid = (NNZ_COUNT * 16 + 255) / 256;        // 300000 blocks

    lgcn_init_kernel<<<vecGrid, 256, 0, stream>>>(user_emb, item_emb, emb0, accb);

    float* cur = emb0;
    float* nxt = emb1;
    for (int layer = 0; layer < N_LAYERS; ++layer) {
        lgcn_zero_kernel<<<vecGrid, 256, 0, stream>>>(nxt);
        lgcn_spmm_kernel<<<spmmGrid, 256, 0, stream>>>(rows, cols, vals, cur, nxt);
        lgcn_acc_kernel<<<vecGrid, 256, 0, stream>>>(accb, nxt);
        float* t = cur; cur = nxt; nxt = t;   // deterministic pointer swap
    }

    // 64 M-tiles x 3125 N-tiles = 200000 waves = 25000 blocks of 8 wave32s
    const int gemmBlocks = (BATCH_USERS / 16) * (ITEMS_COUNT / 16) / 8;
    lgcn_gemm_sigmoid_kernel<<<gemmBlocks, 256, 0, stream>>>(accb, user_ids, out);
}